// ResidualBlockWithPointsBase_16140487098678
// MI455X (gfx1250) — compile-verified
//
#include <hip/hip_runtime.h>

// ---------------------------------------------------------------------------
// ResidualBlockWithPoints for MI455X (gfx1250, wave32, WMMA)
//   layer1: sparse conv (gather-GEMM-scatter) -> BN -> ReLU
//   layer2: sparse conv -> BN -> +x -> ReLU
// N=100000 points, C=64 channels, K=27 offsets (derived from in_sizes).
// GEMM work on V_WMMA_F32_16X16X4_F32 (f32 = reference precision); scatter
// via hardware global_atomic_add_f32. x / h / Y are ~25.6 MB each => fully
// L2-resident (192 MB).
//
// v3: W[k]'s WMMA B-fragments (the whole 64x64 tile = 128 VGPRs/lane) are
// loaded ONCE per wave into registers; each wave then processes 8 M-tiles
// (grid-strided, wave-uniform loop so EXEC stays all-1s for WMMA). This
// removes the 128 VMEM B-loads per tile that made v2 VMEM-issue bound.
// ---------------------------------------------------------------------------

typedef __attribute__((ext_vector_type(2))) float v2f;
typedef __attribute__((ext_vector_type(8))) float v8f;

#define CH 64
#define BN_EPS 1e-5f
#define TPW 8  // tiles per wave (B-register reuse factor)

__device__ __forceinline__ void atomAddF32(float* p, float v) {
  __hip_atomic_fetch_add(p, v, __ATOMIC_RELAXED, __HIP_MEMORY_SCOPE_AGENT);
}

// ---------------------------------------------------------------------------
// Zero fill (b128 stores)
// ---------------------------------------------------------------------------
__global__ void zero_kernel4(float4* __restrict__ p, long long n4) {
  const long long stride = (long long)gridDim.x * blockDim.x;
  for (long long i = (long long)blockIdx.x * blockDim.x + threadIdx.x; i < n4; i += stride)
    p[i] = make_float4(0.f, 0.f, 0.f, 0.f);
}

__global__ void zero_kernel(float* __restrict__ p, long long n) {
  const long long i = (long long)blockIdx.x * blockDim.x + threadIdx.x;
  if (i < n) p[i] = 0.0f;
}

// ---------------------------------------------------------------------------
// Sparse conv. blockIdx.y = k. Each wave: load W[k] B-fragments to registers,
// then loop over M-tiles of 16 gathered rows.
//   A (16x4 f32 per step): lane l -> row l%16, K = 2*(l/16) + {0,1}
//   B (4x16 f32): lane l -> col n0 + l%16, K = 2*(l/16) + {0,1}
//   D (16x16 f32, v8f): lane l, vgpr v -> M = v + 8*(l/16), N = l%16
// ---------------------------------------------------------------------------
__global__ void __launch_bounds__(256) conv_wmma(
    const float* __restrict__ X, const float* __restrict__ W,
    const int* __restrict__ in_idx, const int* __restrict__ out_idx,
    float* __restrict__ Y, int Npts, int tilesPerK, int wavesPerK) {
  const int lane = threadIdx.x & 31;
  const int waveId = blockIdx.x * 8 + (threadIdx.x >> 5);
  if (waveId >= wavesPerK) return;  // wave-uniform exit

  const int k = blockIdx.y;
  const int g = lane >> 4;   // half-wave group (0/1)
  const int r = lane & 15;   // row of A / column of B,D

  // ---- Load all B fragments of W[k] into registers (once per wave) ----
  const float* __restrict__ Wk = W + (long long)k * (CH * CH);
  v2f B[16][4];
#pragma unroll
  for (int kk = 0; kk < 16; ++kk) {
    const int c0 = (kk << 2) + (g << 1);
#pragma unroll
    for (int nt = 0; nt < 4; ++nt) {
      B[kk][nt].x = Wk[c0 * CH + (nt << 4) + r];
      B[kk][nt].y = Wk[(c0 + 1) * CH + (nt << 4) + r];
    }
  }

  const int* __restrict__ iidx_k = in_idx + (long long)k * Npts;
  const int* __restrict__ oidx_k = out_idx + (long long)k * Npts;

  // ---- Grid-strided, wave-uniform tile loop ----
  for (int tile = waveId; tile < tilesPerK; tile += wavesPerK) {
    const int ebase = tile << 4;

    // Gather row index for this lane's A row (tail masked, not predicated,
    // so EXEC remains all-1s through the WMMAs).
    const int er = ebase + r;
    const bool rowValid = (er < Npts);
    const int gi = iidx_k[rowValid ? er : (Npts - 1)];
    const float rowMask = rowValid ? 1.0f : 0.0f;
    const float* __restrict__ xrow = X + (long long)gi * CH;

    v8f acc[4];
#pragma unroll
    for (int nt = 0; nt < 4; ++nt) acc[nt] = (v8f){0, 0, 0, 0, 0, 0, 0, 0};

#pragma unroll
    for (int kk = 0; kk < 16; ++kk) {
      const int c0 = (kk << 2) + (g << 1);
      v2f a;
      a.x = xrow[c0] * rowMask;
      a.y = xrow[c0 + 1] * rowMask;
#pragma unroll
      for (int nt = 0; nt < 4; ++nt) {
        acc[nt] = __builtin_amdgcn_wmma_f32_16x16x4_f32(
            /*neg_a=*/false, a, /*neg_b=*/false, B[kk][nt],
            /*c_mod=*/(short)0, acc[nt], /*reuse_a=*/false, /*reuse_b=*/false);
      }
    }

    // Scatter-add D rows to Y[out_idx]. Divergence here is fine (WMMAs done).
#pragma unroll
    for (int v = 0; v < 8; ++v) {
      const int em = ebase + v + (g << 3);
      if (em < Npts) {
        const int oi = oidx_k[em];
        float* yrow = Y + (long long)oi * CH;
#pragma unroll
        for (int nt = 0; nt < 4; ++nt) {
          atomAddF32(&yrow[(nt << 4) + r], acc[nt][v]);
        }
      }
    }
  }
}

// ---------------------------------------------------------------------------
// Per-channel sum / sumsq over [N, 64]. Row-major => i & 63 is the channel and
// is thread-invariant for a 256-thread block with a 64-multiple grid stride.
// stats[0..63] = sum, stats[64..127] = sumsq.
// ---------------------------------------------------------------------------
__global__ void __launch_bounds__(256) bn_stats(
    const float* __restrict__ Y, float* __restrict__ stats, long long NC) {
  __shared__ float ps[256];
  __shared__ float pq[256];
  const int t = threadIdx.x;
  float s = 0.0f, q = 0.0f;
  const long long stride = (long long)gridDim.x * 256;
  for (long long i = (long long)blockIdx.x * 256 + t; i < NC; i += stride) {
    float v = Y[i];
    s += v;
    q += v * v;
  }
  ps[t] = s;
  pq[t] = q;
  __syncthreads();
  if (t < 64) {
    s = ps[t] + ps[t + 64] + ps[t + 128] + ps[t + 192];
    q = pq[t] + pq[t + 64] + pq[t + 128] + pq[t + 192];
    atomAddF32(&stats[t], s);
    atomAddF32(&stats[64 + t], q);
  }
}

// ---------------------------------------------------------------------------
// BN (batch stats, biased var) + ReLU
// ---------------------------------------------------------------------------
__global__ void __launch_bounds__(256) bn_relu(
    const float* __restrict__ Y, const float* __restrict__ stats,
    const float* __restrict__ gamma, const float* __restrict__ beta,
    float* __restrict__ H, long long NC, float invN) {
  const long long i = (long long)blockIdx.x * blockDim.x + threadIdx.x;
  if (i >= NC) return;
  const int c = (int)(i & (CH - 1));
  const float mu = stats[c] * invN;
  const float var = stats[CH + c] * invN - mu * mu;
  const float h = (Y[i] - mu) * rsqrtf(var + BN_EPS) * gamma[c] + beta[c];
  H[i] = fmaxf(h, 0.0f);
}

// BN + residual + ReLU (final output)
__global__ void __launch_bounds__(256) bn_residual_relu(
    const float* __restrict__ Y, const float* __restrict__ stats,
    const float* __restrict__ gamma, const float* __restrict__ beta,
    const float* __restrict__ X, float* __restrict__ O, long long NC, float invN) {
  const long long i = (long long)blockIdx.x * blockDim.x + threadIdx.x;
  if (i >= NC) return;
  const int c = (int)(i & (CH - 1));
  const float mu = stats[c] * invN;
  const float var = stats[CH + c] * invN - mu * mu;
  const float h = (Y[i] - mu) * rsqrtf(var + BN_EPS) * gamma[c] + beta[c];
  O[i] = fmaxf(h + X[i], 0.0f);
}

// ---------------------------------------------------------------------------
// Launch: x, norm_points(unused), W1, gamma1, beta1, W2, gamma2, beta2,
//         in_idx, out_idx
// ---------------------------------------------------------------------------
extern "C" void kernel_launch(void* const* d_in, const int* in_sizes, int n_in,
                              void* d_out, int out_size, void* d_ws, size_t ws_size,
                              hipStream_t stream) {
  const float* x  = (const float*)d_in[0];
  const float* W1 = (const float*)d_in[2];
  const float* g1 = (const float*)d_in[3];
  const float* b1 = (const float*)d_in[4];
  const float* W2 = (const float*)d_in[5];
  const float* g2 = (const float*)d_in[6];
  const float* b2 = (const float*)d_in[7];
  const int* iidx = (const int*)d_in[8];
  const int* oidx = (const int*)d_in[9];
  float* out = (float*)d_out;

  const int N = in_sizes[0] / CH;           // 100000
  const int K = in_sizes[8] / N;            // 27
  const long long NC = (long long)N * CH;

  float* buf0  = (float*)d_ws;              // conv accumulator (reused)
  float* buf1  = buf0 + NC;                 // hidden activations h
  float* stats = buf1 + NC;                 // 128 floats: sum, sumsq

  const int tilesPerK = (N + 15) / 16;                 // 6250
  const int wavesPerK = (tilesPerK + TPW - 1) / TPW;   // 782
  dim3 convGrid((unsigned)((wavesPerK + 7) / 8), (unsigned)K, 1);
  const int ewBlocks = (int)((NC + 255) / 256);
  const float invN = 1.0f / (float)N;
  const long long NC4 = NC >> 2;  // NC is a multiple of 4 (CH=64)

  // ---- layer 1: conv -> BN -> ReLU ----
  zero_kernel4<<<2048, 256, 0, stream>>>((float4*)buf0, NC4);
  zero_kernel<<<1, 128, 0, stream>>>(stats, 128);
  conv_wmma<<<convGrid, 256, 0, stream>>>(x, W1, iidx, oidx, buf0, N, tilesPerK, wavesPerK);
  bn_stats<<<512, 256, 0, stream>>>(buf0, stats, NC);
  bn_relu<<<ewBlocks, 256, 0, stream>>>(buf0, stats, g1, b1, buf1, NC, invN);

  // ---- layer 2: conv -> BN -> +x -> ReLU ----
  zero_kernel4<<<2048, 256, 0, stream>>>((float4*)buf0, NC4);
  zero_kernel<<<1, 128, 0, stream>>>(stats, 128);
  conv_wmma<<<convGrid, 256, 0, stream>>>(buf1, W2, iidx, oidx, buf0, N, tilesPerK, wavesPerK);
  bn_stats<<<512, 256, 0, stream>>>(buf0, stats, NC);
  bn_residual_relu<<<ewBlocks, 256, 0, stream>>>(buf0, stats, g2, b2, x, out, NC, invN);
}